// VectorQuantizer_790273982755
// MI455X (gfx1250) — compile-verified
//
#include <hip/hip_runtime.h>
#include <hip/hip_bf16.h>

// CDNA5 / gfx1250 VQ-VAE vector quantizer.
// Distances via V_WMMA_F32_16X16X4_F32 (exact f32 — argmin gaps ~1e-5 make
// f16/bf16 unsafe, and the kernel is HBM-bound anyway at 23.3 TB/s).
// Round 2: preload all 16 B fragments per tile so the 8 ds_load_2addr_b64
// issue before a single s_wait_dscnt, letting the 16 WMMAs run back-to-back.

typedef __attribute__((ext_vector_type(2))) float v2f;
typedef __attribute__((ext_vector_type(8))) float v8f;

#define N_ROWS   65536      // 64*32*32 latent vectors
#define DIM      64         // embedding dim
#define KCODES   512        // codebook size
#define CHUNK    128        // codes staged in LDS per pass (32 KB)
#define NCHUNKS  (KCODES / CHUNK)
#define TILES_PER_CHUNK (CHUNK / 16)
#define ROWS_PER_BLOCK 128  // 8 waves * 16 rows
#define GRID_MAIN (N_ROWS / ROWS_PER_BLOCK)

// ---------------------------------------------------------------- ||e_k||^2
__global__ void vq_enorm_kernel(const float* __restrict__ E,
                                float* __restrict__ enorm) {
    int k = blockIdx.x * blockDim.x + threadIdx.x;
    if (k < KCODES) {
        const float* e = E + (size_t)k * DIM;
        float s = 0.0f;
        #pragma unroll
        for (int d = 0; d < DIM; ++d) s += e[d] * e[d];
        enorm[k] = s;
    }
}

// ---------------------------------------------------------------- main
__global__ __launch_bounds__(256) void vq_main_kernel(
    const float* __restrict__ X,      // [N_ROWS, DIM]
    const float* __restrict__ E,      // [KCODES, DIM]
    const float* __restrict__ enorm,  // [KCODES]
    float* __restrict__ outQ,         // [N_ROWS, DIM] (d_out + 1)
    float* __restrict__ partial)      // [GRID_MAIN] loss partials
{
    __shared__ float eLds[CHUNK * DIM];   // 32 KB codebook chunk
    __shared__ float lossWave[8];

    const int tid  = threadIdx.x;
    const int wave = tid >> 5;
    const int lane = tid & 31;
    const int half = lane >> 4;           // 0: lanes 0-15, 1: lanes 16-31
    const int l16  = lane & 15;

    const int rowBase = blockIdx.x * ROWS_PER_BLOCK + wave * 16;

    // A fragments (f32 16x4 per step): lane holds row (l16), K = 4s + 2*half + {0,1}
    v2f aF[16];
    {
        const float* xr = X + (size_t)(rowBase + l16) * DIM + 2 * half;
        #pragma unroll
        for (int s = 0; s < 16; ++s)
            aF[s] = *(const v2f*)(xr + 4 * s);
    }

    float bestS[8];
    int   bestI[8];
    #pragma unroll
    for (int r = 0; r < 8; ++r) { bestS[r] = 3.0e38f; bestI[r] = 0; }

    for (int c = 0; c < NCHUNKS; ++c) {
        __syncthreads();
        // cooperative stage of 128 codes into LDS (2048 float4 / 256 threads)
        {
            const float4* src = (const float4*)(E + (size_t)c * CHUNK * DIM);
            float4* dst = (float4*)eLds;
            #pragma unroll
            for (int i = 0; i < 8; ++i)
                dst[tid + 256 * i] = src[tid + 256 * i];
        }
        __syncthreads();

        for (int t = 0; t < TILES_PER_CHUNK; ++t) {
            const int nLoc  = t * 16;
            const int nGlob = c * CHUNK + nLoc;

            // B fragments (f32 4x16): lane holds code (nLoc + l16), same K
            // pattern as A. Preload all 16 so the DS loads issue as one burst
            // ahead of the WMMA chain (single dscnt wait, no per-pair stall).
            const float* br = eLds + (size_t)(nLoc + l16) * DIM + 2 * half;
            v2f bF[16];
            #pragma unroll
            for (int s = 0; s < 16; ++s)
                bF[s] = *(const v2f*)(br + 4 * s);

            v8f acc = {0.f, 0.f, 0.f, 0.f, 0.f, 0.f, 0.f, 0.f};
            #pragma unroll
            for (int s = 0; s < 16; ++s)
                acc = __builtin_amdgcn_wmma_f32_16x16x4_f32(
                          false, aF[s], false, bF[s], (short)0, acc, false, false);

            // D tile: acc[r] = <row (r + 8*half)> . <code (nGlob + l16)>
            const int   myN = nGlob + l16;
            const float en  = enorm[myN];
            #pragma unroll
            for (int r = 0; r < 8; ++r) {
                float sc = en - 2.0f * acc[r];        // drop ||x||^2 (row const)
                bool better = sc < bestS[r];          // strict: keeps smallest n
                bestI[r] = better ? myN : bestI[r];
                bestS[r] = better ? sc : bestS[r];
            }
        }
    }

    // argmin butterfly across the 16 lanes holding each row (masks stay in half)
    #pragma unroll
    for (int r = 0; r < 8; ++r) {
        float s = bestS[r];
        int   i = bestI[r];
        #pragma unroll
        for (int m = 1; m <= 8; m <<= 1) {
            float so = __shfl_xor(s, m, 32);
            int   io = __shfl_xor(i, m, 32);
            bool take = (so < s) || (so == s && io < i);  // first-occurrence tiebreak
            s = take ? so : s;
            i = take ? io : i;
        }
        bestI[r] = i;
    }

    // gather quantized rows, write out (b32, coalesced), accumulate (q-x)^2
    float localLoss = 0.0f;
    #pragma unroll
    for (int r = 0; r < 8; ++r) {
        const int    row   = rowBase + r + 8 * half;
        const size_t base  = (size_t)row * DIM;
        const size_t ebase = (size_t)bestI[r] * DIM;
        #pragma unroll
        for (int j = 0; j < 4; ++j) {
            const int col = l16 + 16 * j;
            float q = E[ebase + col];
            float x = X[base + col];
            outQ[base + col] = q;
            float d = q - x;
            localLoss += d * d;
        }
    }

    // deterministic loss reduction: wave butterfly -> LDS -> thread 0
    #pragma unroll
    for (int m = 1; m < 32; m <<= 1)
        localLoss += __shfl_xor(localLoss, m, 32);
    if (lane == 0) lossWave[wave] = localLoss;
    __syncthreads();
    if (tid == 0) {
        float s = 0.0f;
        #pragma unroll
        for (int w = 0; w < 8; ++w) s += lossWave[w];
        partial[blockIdx.x] = s;
    }
}

// ---------------------------------------------------------------- finalize
__global__ void vq_finalize_kernel(const float* __restrict__ partial,
                                   float* __restrict__ out) {
    if (threadIdx.x == 0 && blockIdx.x == 0) {
        float s = 0.0f;
        for (int b = 0; b < GRID_MAIN; ++b) s += partial[b];  // fixed order
        out[0] = 1.25f * s / (float)((size_t)N_ROWS * DIM);   // (1 + BETA) * mse
    }
}

extern "C" void kernel_launch(void* const* d_in, const int* in_sizes, int n_in,
                              void* d_out, int out_size, void* d_ws, size_t ws_size,
                              hipStream_t stream) {
    (void)in_sizes; (void)n_in; (void)out_size; (void)ws_size;
    const float* X = (const float*)d_in[0];   // latents  [64,32,32,64]
    const float* E = (const float*)d_in[1];   // embedding [512,64]
    float* out     = (float*)d_out;           // [0]=vq_loss, [1..]=quantized_st

    float* enorm   = (float*)d_ws;            // 512 floats
    float* partial = enorm + KCODES;          // GRID_MAIN floats

    vq_enorm_kernel<<<2, 256, 0, stream>>>(E, enorm);
    vq_main_kernel<<<GRID_MAIN, 256, 0, stream>>>(X, E, enorm, out + 1, partial);
    vq_finalize_kernel<<<1, 32, 0, stream>>>(partial, out);
}